// TripletLoss_11751030522315
// MI455X (gfx1250) — compile-verified
//
#include <hip/hip_runtime.h>
#include <hip/hip_bf16.h>
#include <math.h>

// ---------------------------------------------------------------------------
// Batch-hard triplet loss on MI455X (gfx1250).
//   G = X X^T via V_WMMA_F32_16X16X4_F32 (exact fp32 matrix pipe).
//   - 4 column tiles per wave share one A fragment (5 loads -> 4 WMMAs)
//   - double-buffered fragments to overlap global loads with WMMA
//   - mining done on squared distances (sqrt deferred out of hot loop)
//   - column chunks combined with int-atomic max/min on float bit patterns
// ---------------------------------------------------------------------------

typedef __attribute__((ext_vector_type(2))) float v2f;
typedef __attribute__((ext_vector_type(8))) float v8f;

#define TL_N      4096
#define TL_D      256
#define TL_MARGIN 0.3f
#define TL_EPS    1e-12f
#define TL_CS     8      // column chunks (parallelism across workgroups)
#define TL_CT     4      // column tiles accumulated concurrently per wave

// ---------------------------------------------------------------------------
// Kernel 1: row squared norms + init of the atomic mining buffers.
//   ap2 init = -inf (bits 0xFF800000), an2 init = +inf (bits 0x7F800000).
//   All mined candidates are clamped to >= EPS > 0, so signed-int atomic
//   max/min implements float max/min exactly.
// ---------------------------------------------------------------------------
__global__ void tl_rowsq(const float* __restrict__ X, float* __restrict__ sq,
                         int* __restrict__ ap2_bits, int* __restrict__ an2_bits) {
    int row = blockIdx.x * blockDim.x + threadIdx.x;
    if (row >= TL_N) return;
    const float4* p = (const float4*)(X + row * TL_D);
    float s = 0.f;
#pragma unroll 8
    for (int i = 0; i < TL_D / 4; ++i) {
        float4 v = p[i];
        s = fmaf(v.x, v.x, s);
        s = fmaf(v.y, v.y, s);
        s = fmaf(v.z, v.z, s);
        s = fmaf(v.w, v.w, s);
    }
    sq[row] = s;
    ap2_bits[row] = (int)0xFF800000;  // -inf
    an2_bits[row] = (int)0x7F800000;  // +inf
}

// ---------------------------------------------------------------------------
// Kernel 2: one wave per (16-row tile, column chunk).
// WMMA f32 16x16x4 layouts:
//   A frag: lane L (half h=L>>4, m=L&15): v0 = X[m0+m][k+2h], v1 = next k
//   B frag: identical addressing with column-tile rows (Gram = X * X^T)
//   C/D   : VGPR r -> row (r + 8h), N = L&15
// ---------------------------------------------------------------------------
__global__ void __launch_bounds__(32)
tl_tile(const float* __restrict__ X,
        const int*   __restrict__ labels,
        const float* __restrict__ sq,
        int* __restrict__ ap2_bits,
        int* __restrict__ an2_bits) {
    const int lane = threadIdx.x & 31;
    const int half = lane >> 4;     // 0: lanes 0-15, 1: lanes 16-31
    const int l16  = lane & 15;

    const int m0      = blockIdx.x * 16;             // row tile
    const int n_begin = blockIdx.y * (TL_N / TL_CS); // column chunk
    const int n_end   = n_begin + (TL_N / TL_CS);

    // Per-lane row metadata: slot r covers row m0 + r + 8*half
    float sq_row[8];
    int   lab_row[8];
#pragma unroll
    for (int r = 0; r < 8; ++r) {
        int m = m0 + r + half * 8;
        sq_row[r]  = sq[m];
        lab_row[r] = labels[m];
    }

    const float* pa = X + (m0 + l16) * TL_D + half * 2;

    float apv[8], anv[8];  // running max / min of clamped squared distances
#pragma unroll
    for (int r = 0; r < 8; ++r) { apv[r] = -INFINITY; anv[r] = INFINITY; }

    for (int n0 = n_begin; n0 < n_end; n0 += 16 * TL_CT) {
        const float* pb = X + (n0 + l16) * TL_D + half * 2;

        float sqc[TL_CT];
        int   labc[TL_CT];
#pragma unroll
        for (int t = 0; t < TL_CT; ++t) {
            sqc[t]  = sq[n0 + 16 * t + l16];
            labc[t] = labels[n0 + 16 * t + l16];
        }

        v8f c[TL_CT];
#pragma unroll
        for (int t = 0; t < TL_CT; ++t) c[t] = (v8f){};

        // --- software-pipelined K loop: load k+1 while WMMA-ing k ---
        v2f a = *(const v2f*)pa;
        v2f b[TL_CT];
#pragma unroll
        for (int t = 0; t < TL_CT; ++t)
            b[t] = *(const v2f*)(pb + t * 16 * TL_D);

#pragma unroll 4
        for (int kk = 4; kk < TL_D; kk += 4) {
            v2f a_n = *(const v2f*)(pa + kk);
            v2f b_n[TL_CT];
#pragma unroll
            for (int t = 0; t < TL_CT; ++t)
                b_n[t] = *(const v2f*)(pb + t * 16 * TL_D + kk);
#pragma unroll
            for (int t = 0; t < TL_CT; ++t)
                c[t] = __builtin_amdgcn_wmma_f32_16x16x4_f32(
                           false, a, false, b[t], (short)0, c[t], false, false);
            a = a_n;
#pragma unroll
            for (int t = 0; t < TL_CT; ++t) b[t] = b_n[t];
        }
#pragma unroll
        for (int t = 0; t < TL_CT; ++t)
            c[t] = __builtin_amdgcn_wmma_f32_16x16x4_f32(
                       false, a, false, b[t], (short)0, c[t], false, false);

        // --- fused mining epilogue on squared distances (sqrt deferred) ---
#pragma unroll
        for (int t = 0; t < TL_CT; ++t) {
#pragma unroll
            for (int r = 0; r < 8; ++r) {
                float d2 = fmaf(-2.0f, c[t][r], sq_row[r] + sqc[t]);
                d2 = fmaxf(d2, TL_EPS);            // clamp -> all values > 0
                if (lab_row[r] == labc[t]) {
                    apv[r] = fmaxf(apv[r], d2);    // hardest positive
                } else {
                    anv[r] = fminf(anv[r], d2);    // hardest negative
                }
            }
        }
    }

    // Reduce over N (16 lanes of each half) via xor shuffles 1,2,4,8.
#pragma unroll
    for (int mask = 1; mask <= 8; mask <<= 1) {
#pragma unroll
        for (int r = 0; r < 8; ++r) {
            apv[r] = fmaxf(apv[r], __shfl_xor(apv[r], mask, 32));
            anv[r] = fminf(anv[r], __shfl_xor(anv[r], mask, 32));
        }
    }

    // Combine chunks: signed-int atomics == float max/min for values > 0
    // (-inf / +inf sentinels also order correctly as signed ints).
    if (l16 == 0) {
#pragma unroll
        for (int r = 0; r < 8; ++r) {
            int m = m0 + r + half * 8;
            atomicMax(&ap2_bits[m], __float_as_int(apv[r]));
            atomicMin(&an2_bits[m], __float_as_int(anv[r]));
        }
    }
}

// ---------------------------------------------------------------------------
// Kernel 3: ap = sqrt(ap2), an = sqrt(an2);
//           loss = mean(relu(ap - an + margin)), prec = mean(an > ap)
// ---------------------------------------------------------------------------
__global__ void __launch_bounds__(256)
tl_finalize(const float* __restrict__ ap2,
            const float* __restrict__ an2,
            float* __restrict__ out) {
    __shared__ float s_loss[256];
    __shared__ float s_prec[256];
    const int tid = threadIdx.x;

    float loss = 0.f, prec = 0.f;
    for (int i = tid; i < TL_N; i += 256) {
        float a = sqrtf(ap2[i]);   // values already clamped >= EPS
        float b = sqrtf(an2[i]);
        loss += fmaxf(a - b + TL_MARGIN, 0.f);
        prec += (b > a) ? 1.f : 0.f;
    }
    s_loss[tid] = loss;
    s_prec[tid] = prec;
    __syncthreads();

    for (int s = 128; s > 0; s >>= 1) {
        if (tid < s) {
            s_loss[tid] += s_loss[tid + s];
            s_prec[tid] += s_prec[tid + s];
        }
        __syncthreads();
    }
    if (tid == 0) {
        out[0] = s_loss[0] * (1.0f / TL_N);
        out[1] = s_prec[0] * (1.0f / TL_N);
    }
}

// ---------------------------------------------------------------------------
extern "C" void kernel_launch(void* const* d_in, const int* in_sizes, int n_in,
                              void* d_out, int out_size, void* d_ws, size_t ws_size,
                              hipStream_t stream) {
    const float* X      = (const float*)d_in[0];   // [4096, 256] fp32
    const int*   labels = (const int*)d_in[1];     // [4096] int32
    float*       out    = (float*)d_out;           // [2]: (loss, prec)

    float* sq       = (float*)d_ws;          // [4096]
    int*   ap2_bits = (int*)(sq + TL_N);     // [4096]
    int*   an2_bits = ap2_bits + TL_N;       // [4096]

    tl_rowsq<<<TL_N / 256, 256, 0, stream>>>(X, sq, ap2_bits, an2_bits);

    // (256 row tiles) x (8 column chunks), one wave each
    dim3 grid(TL_N / 16, TL_CS);
    tl_tile<<<grid, 32, 0, stream>>>(X, labels, sq, ap2_bits, an2_bits);

    tl_finalize<<<1, 256, 0, stream>>>((const float*)ap2_bits,
                                       (const float*)an2_bits, out);
}